// RaySamplerPDF_86801289052672
// MI455X (gfx1250) — compile-verified
//
#include <hip/hip_runtime.h>
#include <hip/hip_bf16.h>

// RaySamplerPDF (NeRF sample_pdf) for MI455X / gfx1250.
// Memory-bound: 384 MB traffic -> ~16.5 us floor @ 23.3 TB/s.
// CDF cumsum computed with V_WMMA_F32_16X16X4_F32 against an on-the-fly
// upper-triangular ones matrix; padding normalization applied analytically.

#define NUM_BINS   128
#define NUM_SAMP   128
#define EPSV       1e-5f

#define WAVES_PER_BLOCK 4
#define RAYS_PER_WAVE   16
#define RAYS_PER_BLOCK  (WAVES_PER_BLOCK * RAYS_PER_WAVE)   // 64
#define BLOCK_THREADS   (WAVES_PER_BLOCK * 32)              // 128
#define CSTRIDE         132                                  // 129 used + pad

typedef __attribute__((ext_vector_type(2))) float v2f;
typedef __attribute__((ext_vector_type(8))) float v8f;

__global__ __launch_bounds__(BLOCK_THREADS)
void ray_sampler_pdf_kernel(const float* __restrict__ weights,
                            const float* __restrict__ u,
                            const int*   __restrict__ strat_ptr,
                            float*       __restrict__ out)
{
  // Single LDS buffer: staged raw weights live in columns 0..127 of each row,
  // WMMA writes the raw cumsum into columns 1..128 (descending tile order makes
  // the overlap safe: each tile reads cols <= 16t+15 before writing 16t+1..16t+16,
  // and DS ops of a wave execute in order).
  __shared__ __align__(16) float cdfF[RAYS_PER_BLOCK][CSTRIDE];

  const int tid  = threadIdx.x;
  const int lane = tid & 31;
  const int wave = tid >> 5;
  const int blockRay0 = blockIdx.x * RAYS_PER_BLOCK;
  const int waveRay0  = wave * RAYS_PER_WAVE;          // local row base
  const bool strat = (strat_ptr[0] != 0);

  // ---- Phase 1: stage weights, fully coalesced float4 loads -------------
  {
    const float4* gsrc = (const float4*)(weights + (size_t)blockRay0 * NUM_BINS);
    #pragma unroll 4
    for (int i = tid; i < RAYS_PER_BLOCK * NUM_BINS / 4; i += BLOCK_THREADS) {
      float4 w4 = gsrc[i];
      int r  = i >> 5;            // 32 float4 per 128-float row
      int c4 = (i & 31) << 2;
      *(float4*)&cdfF[r][c4] = w4;
    }
  }
  __syncthreads();

  // ---- Phase 2: raw cumsum via WMMA (D = A[16x128] * UpperTriOnes) ------
  {
    const int n    = lane & 15;          // column within tile / A row
    const int half = lane >> 4;          // 0: K offset 0,1   1: K offset 2,3
    const int koff = half << 1;
    const float* arow = &cdfF[waveRay0 + n][0];

    for (int t = 7; t >= 0; --t) {       // descending: see aliasing note above
      v8f acc = {0.f,0.f,0.f,0.f,0.f,0.f,0.f,0.f};
      const int col = t * 16 + n;        // global output bin for this lane
      const int cmax = 4 * t + 3;        // last chunk intersecting this tile
      for (int c = 0; c <= cmax; ++c) {
        const int kb = 4 * c + koff;
        v2f a = *(const v2f*)&arow[kb];  // 8B-aligned LDS load
        v2f b;
        b.x = (kb     <= col) ? 1.0f : 0.0f;   // triangular ones matrix
        b.y = (kb + 1 <= col) ? 1.0f : 0.0f;
        acc = __builtin_amdgcn_wmma_f32_16x16x4_f32(
            /*neg_a=*/false, a, /*neg_b=*/false, b,
            /*c_mod=*/(short)0, acc, /*reuse_a=*/false, /*reuse_b=*/false);
      }
      // C/D layout: VGPR g holds rows g (lanes 0-15) and g+8 (lanes 16-31)
      const int mrow = waveRay0 + half * 8;
      #pragma unroll
      for (int g = 0; g < 8; ++g)
        cdfF[mrow + g][1 + t * 16 + n] = acc[g];
    }
  }
  __syncthreads();

  // ---- Phase 3: normalization with analytic padding correction ----------
  // cdf[j] = (rawcum[j] + padding*j/128) / (sum + padding),  cdf[0] = 0
  for (int m = 0; m < RAYS_PER_WAVE; ++m) {
    float* crow   = cdfF[waveRay0 + m];
    float sumraw  = crow[NUM_BINS];                  // rawcum[128] = total
    float padding = fmaxf(EPSV - sumraw, 0.0f);
    float inv     = 1.0f / (sumraw + padding);
    float padstep = padding * (1.0f / NUM_BINS);
    for (int j = lane; j <= NUM_BINS; j += 32)
      crow[j] = (j == 0) ? 0.0f : (crow[j] + padstep * (float)j) * inv;
  }
  __syncthreads();

  // ---- Phase 4: inverse-CDF sampling (branchless 8-probe binary search) -
  {
    const float invS = 1.0f / NUM_SAMP;
    for (int f = lane; f < RAYS_PER_WAVE * NUM_SAMP; f += 32) {
      const int m = f >> 7;              // local ray
      const int s = f & (NUM_SAMP - 1);  // sample index
      const size_t gidx = (size_t)(blockRay0 + waveRay0 + m) * NUM_SAMP + s;
      const float uu = u[gidx];
      const float us = strat ? ((float)s + uu)   * invS
                             : ((float)s + 0.5f) * invS;
      const float* crow = cdfF[waveRay0 + m];
      // largest pos in [0,128] with crow[pos] <= us  (crow[0]==0 <= us always)
      int pos = 0;
      #pragma unroll
      for (int step = 128; step >= 1; step >>= 1) {
        const int np = pos + step;
        if (np <= NUM_BINS && crow[np] <= us) pos = np;
      }
      const int below = pos;                               // = clip(ind-1,0,128)
      const int above = (pos + 1 < NUM_BINS) ? pos + 1 : NUM_BINS;
      const float cb = crow[below];
      const float ca = crow[above];
      const float bb = (float)below * (1.0f / NUM_BINS);   // linspace(0,1,129)
      const float ba = (float)above * (1.0f / NUM_BINS);
      float denom = ca - cb;
      denom = (denom < 1e-5f) ? 1.0f : denom;
      const float tt = (us - cb) / denom;
      out[gidx] = bb + tt * (ba - bb);
    }
  }
}

extern "C" void kernel_launch(void* const* d_in, const int* in_sizes, int n_in,
                              void* d_out, int out_size, void* d_ws, size_t ws_size,
                              hipStream_t stream) {
  const float* weights = (const float*)d_in[0];
  const float* u       = (const float*)d_in[1];
  const int*   strat   = (const int*)d_in[2];
  float*       out     = (float*)d_out;

  const int num_rays = in_sizes[0] / NUM_BINS;        // 262144
  const int grid     = num_rays / RAYS_PER_BLOCK;     // 4096

  ray_sampler_pdf_kernel<<<grid, BLOCK_THREADS, 0, stream>>>(weights, u, strat, out);
}